// GATLayer_6124623364495
// MI455X (gfx1250) — compile-verified
//
#include <hip/hip_runtime.h>

// GAT forward: B=8, N=1024, F_in=128, H=4, Dh=64
// out = [output (B*N*H*Dh), attention (B*H*N*N)] fp32, concatenated flat.

#define B_   8
#define N_   1024
#define FIN_ 128
#define H_   4
#define DH_  64
#define NEG_SLOPE_ 0.2f
#define PS_  1027   // padded LDS stride for score tile (1027 % 64 = 3 -> conflict-free)

typedef __attribute__((ext_vector_type(2))) float v2f;
typedef __attribute__((ext_vector_type(8))) float v8f;

__device__ __forceinline__ v8f wmma4_f32(v2f a, v2f b, v8f c) {
    // V_WMMA_F32_16X16X4_F32 : exact fp32 MMA, D = A(16x4) * B(4x16) + C
    return __builtin_amdgcn_wmma_f32_16x16x4_f32(
        /*neg_a=*/false, a, /*neg_b=*/false, b,
        /*c_mod=*/(short)0, c, /*reuse_a=*/false, /*reuse_b=*/false);
}

// ---------------------------------------------------------------------------
// Kernel 1: h[b,h,n,d] = sum_i x[b,n,i] * W[h,i,d]   (fp32 WMMA, K steps of 4)
// grid = B*H*(N/16) blocks, 128 threads (4 waves, one 16-wide d-tile each)
// ---------------------------------------------------------------------------
__global__ __launch_bounds__(128)
void gat_project(const float* __restrict__ x, const float* __restrict__ W,
                 float* __restrict__ h) {
    const int nt   = blockIdx.x & 63;
    const int bhd  = blockIdx.x >> 6;           // b*H + hd
    const int hd   = bhd & (H_ - 1);
    const int b    = bhd >> 2;
    const int n0   = nt * 16;
    const int wv   = threadIdx.x >> 5;          // d-tile 0..3
    const int lane = threadIdx.x & 31;
    const int row  = lane & 15;                 // A: M index / B,D: N index
    const int half = lane >> 4;                 // K-pair select / D row-half

    const float* xrow  = x + ((size_t)b * N_ + n0) * FIN_;
    const float* wbase = W + (size_t)hd * FIN_ * DH_;
    const int d = wv * 16 + row;

    v8f acc = {};
    for (int k = 0; k < FIN_; k += 4) {
        const int kk = k + half * 2;
        v2f av, bv;
        av.x = xrow[(size_t)row * FIN_ + kk];
        av.y = xrow[(size_t)row * FIN_ + kk + 1];
        bv.x = wbase[(size_t)kk * DH_ + d];
        bv.y = wbase[(size_t)(kk + 1) * DH_ + d];
        acc = wmma4_f32(av, bv, acc);
    }

    float* hout = h + ((size_t)bhd * N_ + n0) * DH_;
#pragma unroll
    for (int i = 0; i < 8; ++i) {
        const int m = half * 8 + i;             // D layout: VGPR i -> M=i / M=8+i
        hout[(size_t)m * DH_ + wv * 16 + (lane & 15)] = acc[i];
    }
}

// ---------------------------------------------------------------------------
// Kernel 2: e_src / e_dst dot products (negligible cost)
// ---------------------------------------------------------------------------
__global__ __launch_bounds__(256)
void gat_edge(const float* __restrict__ h, const float* __restrict__ a,
              float* __restrict__ esrc, float* __restrict__ edst) {
    const int t = blockIdx.x * 256 + threadIdx.x;    // 0 .. B*H*N-1
    const int n   = t & (N_ - 1);
    const int bhd = t >> 10;
    const int hd  = bhd & (H_ - 1);
    const float* hrow = h + ((size_t)bhd * N_ + n) * DH_;
    const float* asrc = a + (size_t)hd * 2 * DH_;
    const float* adst = asrc + DH_;
    float s1 = 0.f, s2 = 0.f;
#pragma unroll 8
    for (int d = 0; d < DH_; ++d) {
        const float hv = hrow[d];
        s1 += hv * asrc[d];
        s2 += hv * adst[d];
    }
    esrc[t] = s1;
    edst[t] = s2;
}

// ---------------------------------------------------------------------------
// Kernel 3: scores -> masked leaky-relu -> softmax -> write attention (NT),
//           then h' = attn @ h with fp32 WMMA straight out of LDS.
// grid = B*H*(N/16) blocks, 256 threads (8 waves). LDS ~79 KB (gfx1250: 320 KB).
// ---------------------------------------------------------------------------
__global__ __launch_bounds__(256)
void gat_attend(const float* __restrict__ h, const int* __restrict__ adj,
                const float* __restrict__ esrc, const float* __restrict__ edst,
                float* __restrict__ out, float* __restrict__ attn) {
    __shared__ float p[16 * PS_];        // 16 rows x 1024 scores (padded)
    __shared__ float edst_s[N_];
    __shared__ float esrc_s[16];
    __shared__ float red[16][16];
    __shared__ float rowmax_s[16];
    __shared__ float rowsum_s[16];
    __shared__ float part[8 * 256];      // cross-wave h' partials

    const int nt  = blockIdx.x & 63;
    const int bhd = blockIdx.x >> 6;
    const int hd  = bhd & (H_ - 1);
    const int b   = bhd >> 2;
    const int n0  = nt * 16;
    const int t   = threadIdx.x;

    for (int i = t; i < N_; i += 256) edst_s[i] = edst[(size_t)bhd * N_ + i];
    if (t < 16) esrc_s[t] = esrc[(size_t)bhd * N_ + n0 + t];
    __syncthreads();

    const int r = t >> 4;                // row 0..15
    const int c = t & 15;                // 16 threads per row
    const int* adjrow = adj + (size_t)(n0 + r) * N_;
    const float er = esrc_s[r];

    // Phase A: masked leaky-relu scores + row max
    float lmax = -3.0e38f;
    for (int j = 0; j < 64; ++j) {
        const int m = c + (j << 4);
        const float e = er + edst_s[m];
        const float s = (e > 0.f) ? e : (NEG_SLOPE_ * e);
        const float val = (adjrow[m] == 0) ? -1.0e30f : s;
        p[r * PS_ + m] = val;
        lmax = fmaxf(lmax, val);
    }
    red[r][c] = lmax;
    __syncthreads();
    if (c == 0) {
        float mx = red[r][0];
#pragma unroll
        for (int i = 1; i < 16; ++i) mx = fmaxf(mx, red[r][i]);
        rowmax_s[r] = mx;
    }
    __syncthreads();

    // Phase B: exp + row sum
    const float mxr = rowmax_s[r];
    float lsum = 0.f;
    for (int j = 0; j < 64; ++j) {
        const int m = c + (j << 4);
        const float val = p[r * PS_ + m];
        const float ex = (val < -1.0e29f) ? 0.f : __expf(val - mxr);
        p[r * PS_ + m] = ex;
        lsum += ex;
    }
    red[r][c] = lsum;
    __syncthreads();
    if (c == 0) {
        float sm = 0.f;
#pragma unroll
        for (int i = 0; i < 16; ++i) sm += red[r][i];
        rowsum_s[r] = sm;
    }
    __syncthreads();

    // Phase C: normalize; single coalesced streaming (non-temporal) HBM write
    // of the attention tensor. It is never re-read (LDS copy feeds the WMMA),
    // so NT keeps h/adj resident in the 192 MB L2 across blocks.
    const float inv = 1.0f / rowsum_s[r];
    float* arow = attn + ((size_t)bhd * N_ + (n0 + r)) * N_;
    for (int j = 0; j < 64; ++j) {
        const int m = c + (j << 4);
        const float aval = p[r * PS_ + m] * inv;
        p[r * PS_ + m] = aval;
        __builtin_nontemporal_store(aval, &arow[m]);
    }
    __syncthreads();

    // Phase D: h'(16 x 64) = attn(16 x 1024) @ h(1024 x 64) via fp32 WMMA.
    // wave w: d-tile = w&3, K-half = w>>2 (512 each), 128 WMMAs per wave.
    const int w    = t >> 5;
    const int lane = t & 31;
    const int dt   = w & 3;
    const int kh   = w >> 2;
    const int row  = lane & 15;
    const int half = lane >> 4;
    const float* hb = h + (size_t)bhd * N_ * DH_;

    v8f acc = {};
    const int k0 = kh * 512;
    for (int k = k0; k < k0 + 512; k += 4) {
        const int kk = k + half * 2;
        v2f av, bv;
        av.x = p[row * PS_ + kk];                    // A from LDS (conflict-free)
        av.y = p[row * PS_ + kk + 1];
        bv.x = hb[(size_t)kk * DH_ + dt * 16 + row]; // B from L2-resident h
        bv.y = hb[(size_t)(kk + 1) * DH_ + dt * 16 + row];
        acc = wmma4_f32(av, bv, acc);
    }

    float* pw = part + (w << 8) + (lane << 3);
#pragma unroll
    for (int i = 0; i < 8; ++i) pw[i] = acc[i];
    __syncthreads();

    if (kh == 0) {
        const float* q = part + ((w + 4) << 8) + (lane << 3);
#pragma unroll
        for (int i = 0; i < 8; ++i) {
            const float v = acc[i] + q[i];
            const int mrow = half * 8 + i;
            __builtin_nontemporal_store(
                v, &out[((size_t)b * N_ + (n0 + mrow)) * (H_ * DH_) + hd * DH_ +
                        dt * 16 + (lane & 15)]);
        }
    }
}

// ---------------------------------------------------------------------------
extern "C" void kernel_launch(void* const* d_in, const int* in_sizes, int n_in,
                              void* d_out, int out_size, void* d_ws, size_t ws_size,
                              hipStream_t stream) {
    const float* x   = (const float*)d_in[0];   // (B,N,F_in) fp32
    const int*   adj = (const int*)  d_in[1];   // (N,N) int32
    const float* W   = (const float*)d_in[2];   // (H,F_in,Dh) fp32
    const float* a   = (const float*)d_in[3];   // (H,2*Dh,1) fp32

    float* out  = (float*)d_out;                            // (B,N,H*Dh)
    float* attn = out + (size_t)B_ * N_ * H_ * DH_;         // (B,H,N,N)

    float* h    = (float*)d_ws;                             // (B*H,N,Dh) 8 MB
    float* esrc = h + (size_t)B_ * H_ * N_ * DH_;           // (B*H,N)
    float* edst = esrc + (size_t)B_ * H_ * N_;              // (B*H,N)

    gat_project<<<dim3(B_ * H_ * (N_ / 16)), dim3(128), 0, stream>>>(x, W, h);
    gat_edge<<<dim3((B_ * H_ * N_) / 256), dim3(256), 0, stream>>>(h, a, esrc, edst);
    gat_attend<<<dim3(B_ * H_ * (N_ / 16)), dim3(256), 0, stream>>>(h, adj, esrc, edst, out, attn);
}